// AlternateConvolution_3762391351853
// MI455X (gfx1250) — compile-verified
//
#include <hip/hip_runtime.h>

typedef __attribute__((ext_vector_type(2))) float v2f;
typedef __attribute__((ext_vector_type(8))) float v8f;

#define NVV 2048
#define NEE 8192
#define INV 128
#define OUTV 128

// ---------------------------------------------------------------------------
// Kernel 1: w[e] = dot(H_e[e,:], p)   (NE x 128) . (128)
// ---------------------------------------------------------------------------
__global__ __launch_bounds__(256) void k_edge_w(const float* __restrict__ He,
                                                const float* __restrict__ p,
                                                float* __restrict__ w) {
    int e = blockIdx.x * blockDim.x + threadIdx.x;
    if (e >= NEE) return;
    const float4* he = (const float4*)(He + (size_t)e * INV);
    const float4* pp = (const float4*)p;
    float acc = 0.0f;
#pragma unroll
    for (int i = 0; i < INV / 4; ++i) {
        float4 a = he[i], b = pp[i];
        acc += a.x * b.x + a.y * b.y + a.z * b.z + a.w * b.w;
    }
    w[e] = acc;
}

// ---------------------------------------------------------------------------
// Kernel 2: HW = H_v @ weight   (2048x128)x(128x128), one 16x16 tile per wave
// ---------------------------------------------------------------------------
__global__ __launch_bounds__(256) void k_hw(const float* __restrict__ Hv,
                                            const float* __restrict__ Wt,
                                            float* __restrict__ HW) {
    int lane = threadIdx.x & 31;
    int wv   = threadIdx.x >> 5;
    int tile = blockIdx.x * 8 + wv;            // 1024 tiles total
    int tm = tile >> 3;                        // 128 row tiles
    int tn = tile & 7;                         // 8 col tiles
    int m = lane & 15, half = lane >> 4;

    v8f acc = {};
    const float* Arow = Hv + (size_t)(tm * 16 + m) * INV + 2 * half;
    const float* Bcol = Wt + tn * 16 + m;
#pragma unroll 8
    for (int k = 0; k < INV; k += 4) {
        v2f a = *(const v2f*)(Arow + k);
        int kk = k + 2 * half;
        v2f b;
        b.x = Bcol[(size_t)kk * OUTV];
        b.y = Bcol[(size_t)(kk + 1) * OUTV];
        acc = __builtin_amdgcn_wmma_f32_16x16x4_f32(false, a, false, b,
                                                    (short)0, acc, false, false);
    }
#pragma unroll
    for (int r = 0; r < 8; ++r)
        HW[(size_t)(tm * 16 + r + 8 * half) * OUTV + tn * 16 + m] = acc[r];
}

// ---------------------------------------------------------------------------
// Kernel 3: adjA = (eye + (1-eye) * ((T*w) @ T^T)) * adj_v
// Block tile 128x64. 8 waves arranged 4x2; each wave computes 32x32 = 2x2
// WMMA tiles -> 4 frag loads feed 4 WMMAs per K=4 step (1.0 ds-load/WMMA).
// K staged through LDS in 64-chunks; A panel pre-scaled by w[k].
// LDS row stride 68 floats -> conflict-free b64 fragment reads.
// ---------------------------------------------------------------------------
#define KC  64
#define LDP 68
__global__ __launch_bounds__(256) void k_mult(const float* __restrict__ T,
                                              const float* __restrict__ w,
                                              const float* __restrict__ adjv,
                                              float* __restrict__ adjA) {
    __shared__ float Ap[128 * LDP];   // 34816 B
    __shared__ float Bp[64 * LDP];    // 17408 B

    const int i0 = blockIdx.y * 128;
    const int j0 = blockIdx.x * 64;
    const int tid  = threadIdx.x;
    const int lane = tid & 31;
    const int wv   = tid >> 5;
    const int wr = wv >> 1;          // 0..3  -> 32-row slab
    const int wc = wv & 1;           // 0..1  -> 32-col slab
    const int m = lane & 15, half = lane >> 4;

    // cooperative panel-load mapping
    const int arow = tid >> 1;        // 0..127, 32 floats (8 float4) each
    const int aseg = (tid & 1) * 32;  // 0 or 32
    const int brow = tid >> 2;        // 0..63, 16 floats (4 float4) each
    const int bseg = (tid & 3) * 16;  // 0,16,32,48

    v8f acc00 = {}, acc01 = {}, acc10 = {}, acc11 = {};

    for (int k0 = 0; k0 < NEE; k0 += KC) {
        // A panel: rows i0..i0+127, scaled by w[k]
        {
            const float* gA = T + (size_t)(i0 + arow) * NEE + k0 + aseg;
            float* sA = Ap + arow * LDP + aseg;
#pragma unroll
            for (int q = 0; q < 8; ++q) {
                float4 a  = *(const float4*)(gA + 4 * q);
                float4 ww = *(const float4*)(w + k0 + aseg + 4 * q);
                a.x *= ww.x; a.y *= ww.y; a.z *= ww.z; a.w *= ww.w;
                *(float4*)(sA + 4 * q) = a;
            }
        }
        // B panel: rows j0..j0+63 (unscaled)
        {
            const float* gB = T + (size_t)(j0 + brow) * NEE + k0 + bseg;
            float* sB = Bp + brow * LDP + bseg;
#pragma unroll
            for (int q = 0; q < 4; ++q)
                *(float4*)(sB + 4 * q) = *(const float4*)(gB + 4 * q);
        }
        __syncthreads();

        const float* aB0 = Ap + (wr * 32 + m) * LDP + 2 * half;
        const float* aB1 = Ap + (wr * 32 + 16 + m) * LDP + 2 * half;
        const float* bB0 = Bp + (wc * 32 + m) * LDP + 2 * half;
        const float* bB1 = Bp + (wc * 32 + 16 + m) * LDP + 2 * half;
#pragma unroll
        for (int k = 0; k < KC; k += 4) {
            v2f a0 = *(const v2f*)(aB0 + k);
            v2f a1 = *(const v2f*)(aB1 + k);
            v2f b0 = *(const v2f*)(bB0 + k);
            v2f b1 = *(const v2f*)(bB1 + k);
            acc00 = __builtin_amdgcn_wmma_f32_16x16x4_f32(false, a0, false, b0,
                                                          (short)0, acc00, false, false);
            acc01 = __builtin_amdgcn_wmma_f32_16x16x4_f32(false, a0, false, b1,
                                                          (short)0, acc01, false, false);
            acc10 = __builtin_amdgcn_wmma_f32_16x16x4_f32(false, a1, false, b0,
                                                          (short)0, acc10, false, false);
            acc11 = __builtin_amdgcn_wmma_f32_16x16x4_f32(false, a1, false, b1,
                                                          (short)0, acc11, false, false);
        }
        __syncthreads();
    }

    // epilogue: diag -> 1, then mask by adj_v  (4 tiles of 16x16)
    const int ci  = i0 + wr * 32;
    const int cj  = j0 + wc * 32;
#pragma unroll
    for (int r = 0; r < 8; ++r) {
        int row0 = ci + r + 8 * half;        // tiles (0,*)
        int row1 = ci + 16 + r + 8 * half;   // tiles (1,*)
        int col0 = cj + m;                   // tiles (*,0)
        int col1 = cj + 16 + m;              // tiles (*,1)
        float v00 = (row0 == col0) ? 1.0f : acc00[r];
        float v01 = (row0 == col1) ? 1.0f : acc01[r];
        float v10 = (row1 == col0) ? 1.0f : acc10[r];
        float v11 = (row1 == col1) ? 1.0f : acc11[r];
        adjA[(size_t)row0 * NVV + col0] = v00 * adjv[(size_t)row0 * NVV + col0];
        adjA[(size_t)row0 * NVV + col1] = v01 * adjv[(size_t)row0 * NVV + col1];
        adjA[(size_t)row1 * NVV + col0] = v10 * adjv[(size_t)row1 * NVV + col0];
        adjA[(size_t)row1 * NVV + col1] = v11 * adjv[(size_t)row1 * NVV + col1];
    }
}

// ---------------------------------------------------------------------------
// Kernel 4: ret = adjA @ HW + bias   (2048x2048)x(2048x128); L2-resident K loop
// ---------------------------------------------------------------------------
__global__ __launch_bounds__(256) void k_out(const float* __restrict__ adjA,
                                             const float* __restrict__ HW,
                                             const float* __restrict__ bias,
                                             float* __restrict__ ret) {
    int lane = threadIdx.x & 31;
    int wv   = threadIdx.x >> 5;
    int tile = blockIdx.x * 8 + wv;
    int tm = tile >> 3;
    int tn = tile & 7;
    int m = lane & 15, half = lane >> 4;

    v8f acc = {};
    const float* Arow = adjA + (size_t)(tm * 16 + m) * NVV + 2 * half;
    const float* Bcol = HW + tn * 16 + m;
#pragma unroll 4
    for (int k = 0; k < NVV; k += 4) {
        v2f a = *(const v2f*)(Arow + k);
        int kk = k + 2 * half;
        v2f b;
        b.x = Bcol[(size_t)kk * OUTV];
        b.y = Bcol[(size_t)(kk + 1) * OUTV];
        acc = __builtin_amdgcn_wmma_f32_16x16x4_f32(false, a, false, b,
                                                    (short)0, acc, false, false);
    }
    float bb = bias[tn * 16 + m];
#pragma unroll
    for (int r = 0; r < 8; ++r)
        ret[(size_t)(tm * 16 + r + 8 * half) * OUTV + tn * 16 + m] = acc[r] + bb;
}

// ---------------------------------------------------------------------------
extern "C" void kernel_launch(void* const* d_in, const int* in_sizes, int n_in,
                              void* d_out, int out_size, void* d_ws, size_t ws_size,
                              hipStream_t stream) {
    const float* Hv   = (const float*)d_in[0];   // [2048,128]
    const float* He   = (const float*)d_in[1];   // [8192,128]
    // d_in[2] = adj_e (unused by node_layer branch)
    const float* adjv = (const float*)d_in[3];   // [2048,2048]
    const float* T    = (const float*)d_in[4];   // [2048,8192]
    const float* Wt   = (const float*)d_in[5];   // [128,128]
    const float* p    = (const float*)d_in[6];   // [1,128]
    const float* bias = (const float*)d_in[7];   // [128]

    float* out_ret = (float*)d_out;                         // [2048,128]
    float* out_He  = (float*)d_out + (size_t)NVV * OUTV;    // [8192,128] passthrough

    // workspace layout (aligned slices)
    char* ws = (char*)d_ws;
    float* w_vec = (float*)ws;                              // NE floats (32 KB)
    float* HW    = (float*)(ws + 64 * 1024);                // NV*OUTV floats (1 MB)
    float* adjA  = (float*)(ws + 64 * 1024 + 2 * 1024 * 1024); // NV*NV floats (16.8 MB)

    // 1) edge scalar weights
    k_edge_w<<<NEE / 256, 256, 0, stream>>>(He, p, w_vec);

    // 2) HW = H_v @ weight  (1024 tiles / 8 waves per block)
    k_hw<<<(NVV / 16) * (OUTV / 16) / 8, 256, 0, stream>>>(Hv, Wt, HW);

    // 3) adjA = mask(diag->1, (T*w)@T^T) * adj_v   -- the 68.7 GFLOP GEMM
    dim3 grid3(NVV / 64, NVV / 128);   // 32 x 16 = 512 blocks
    k_mult<<<grid3, 256, 0, stream>>>(T, w_vec, adjv, adjA);

    // 4) ret = adjA @ HW + bias
    k_out<<<(NVV / 16) * (OUTV / 16) / 8, 256, 0, stream>>>(adjA, HW, bias, out_ret);

    // 5) second tuple element: H_e passthrough
    hipMemcpyAsync(out_He, He, (size_t)NEE * INV * sizeof(float),
                   hipMemcpyDeviceToDevice, stream);
}